// kNNRepulsionLoss_70265664962783
// MI455X (gfx1250) — compile-verified
//
#include <hip/hip_runtime.h>

// ---------------------------------------------------------------------------
// kNN repulsion loss on MI455X (gfx1250, wave32).
//   B=512 batches, N=2048 points, C=3, S=20 seeds (FPS), K=10 neighbors,
//   H=0.01 -> 1/H^2 = 1e4.  Output: scalar f32.
//
// Design: tiny (12.6MB) latency-bound problem -> keep f32, use
// V_WMMA_F32_16X16X4_F32 for the [N x S] distance GEMM (K=3 padded to 4,
// with the 4th K slot folding in the -2 factor and |s|^2), and exploit the
// 320KB WGP LDS to keep a full 16-seed x 2048-point d^2 slab resident for
// the top-11 selection.  Wave-level loops are driven by readfirstlane'd
// (SGPR) trip counts so EXEC is provably all-ones around every WMMA.
// ---------------------------------------------------------------------------

#define B_    512
#define N_    2048
#define S_    20
#define KNN   10
#define HINV2 10000.0f
#define BIGF  3.0e38f

typedef __attribute__((ext_vector_type(2))) float v2f;
typedef __attribute__((ext_vector_type(8))) float v8f;

// --------------------------- init output -----------------------------------
__global__ void krl_zero_out(float* out) {
  if (threadIdx.x == 0 && blockIdx.x == 0) out[0] = 0.0f;
}

// --------------------------- FPS (one block per batch) ----------------------
__global__ __launch_bounds__(256) void krl_fps_kernel(
    const float* __restrict__ pcs, float* __restrict__ seeds) {
  __shared__ float sx[N_], sy[N_], sz[N_], sd[N_];
  __shared__ float rv[256];
  __shared__ int   ri[256];
  __shared__ int   sfar;

  const int b   = blockIdx.x;
  const int tid = threadIdx.x;
  const float* p = pcs + (size_t)b * N_ * 3;

  for (int i = tid; i < N_; i += 256) {
    sx[i] = p[3 * i + 0];
    sy[i] = p[3 * i + 1];
    sz[i] = p[3 * i + 2];
    sd[i] = 1.0e10f;                       // reference init: 1e10
  }
  __syncthreads();

  int far = 0;
  for (int s = 0; s < S_; ++s) {
    const float cx = sx[far], cy = sy[far], cz = sz[far];
    if (tid == 0) {                        // seed s = centroid used this iter
      float* o = seeds + ((size_t)b * S_ + s) * 3;
      o[0] = cx; o[1] = cy; o[2] = cz;
    }
    float lv = -1.0f; int li = 0;
    for (int i = tid; i < N_; i += 256) {
      const float dx = sx[i] - cx, dy = sy[i] - cy, dz = sz[i] - cz;
      const float d  = dx * dx + dy * dy + dz * dz;
      const float nd = fminf(sd[i], d);
      sd[i] = nd;
      if (nd > lv) { lv = nd; li = i; }    // strided -> keeps first occurrence
    }
    rv[tid] = lv; ri[tid] = li;
    __syncthreads();
    for (int off = 128; off > 0; off >>= 1) {
      if (tid < off) {
        const float ov = rv[tid + off];
        const int   oi = ri[tid + off];
        // jnp.argmax tie-break: first (lowest) index wins
        if (ov > rv[tid] || (ov == rv[tid] && oi < ri[tid])) {
          rv[tid] = ov; ri[tid] = oi;
        }
      }
      __syncthreads();
    }
    if (tid == 0) sfar = ri[0];
    __syncthreads();
    far = sfar;
  }
}

// ---------------- distance WMMA + top-11 select + reduce --------------------
// grid = B_ * 2 blocks; block (b, stile) handles seeds [stile*16, stile*16+16)
__global__ __launch_bounds__(256) void krl_knn_repulsion_kernel(
    const float* __restrict__ pcs, const float* __restrict__ seeds,
    float* __restrict__ out) {
  // stride 2052 (multiple of 4 floats) -> per-lane 4-dword runs are 16B
  // aligned so the 8 D-values merge into two ds_store_b128.
  __shared__ float dist[16][2052];         // ~131 KB of the 320 KB WGP LDS
  __shared__ float psq[N_];                // |p|^2
  __shared__ float sbx[16], sby[16], sbz[16], sbq[16];

  const int blk   = blockIdx.x;
  const int b     = blk >> 1;
  const int stile = blk & 1;
  const int tid   = threadIdx.x;
  const int lane  = tid & 31;
  // Wave id as an SGPR value: loops bounded by it are scalar-controlled, so
  // EXEC stays all-ones across the WMMA (ISA 7.12 requirement).
  const int wid   = __builtin_amdgcn_readfirstlane(tid >> 5);
  const int nvalid = stile ? (S_ - 16) : 16;   // 16 or 4 real seeds

  const float* p = pcs + (size_t)b * N_ * 3;

  // seed tile (pad missing seeds with zeros; their rows are never selected)
  if (tid < 16) {
    const int s = stile * 16 + tid;
    if (s < S_) {
      const float* sp = seeds + ((size_t)b * S_ + s) * 3;
      const float x = sp[0], y = sp[1], z = sp[2];
      sbx[tid] = x; sby[tid] = y; sbz[tid] = z; sbq[tid] = x * x + y * y + z * z;
    } else {
      sbx[tid] = 0.f; sby[tid] = 0.f; sbz[tid] = 0.f; sbq[tid] = 0.f;
    }
  }
  for (int i = tid; i < N_; i += 256) {
    const float x = p[3 * i + 0], y = p[3 * i + 1], z = p[3 * i + 2];
    psq[i] = x * x + y * y + z * z;
  }
  __syncthreads();

  // B operand (4x16 f32, ISA 7.12.2): row K striped across lanes per VGPR.
  //   lanes 0-15 : {K0 = -2*sx[N], K1 = -2*sy[N]}
  //   lanes 16-31: {K2 = -2*sz[N], K3 = |s|^2 }   (A's K3 element is 1)
  const int scol = lane & 15;
  v2f bop;
  if (lane < 16) { bop.x = -2.0f * sbx[scol]; bop.y = -2.0f * sby[scol]; }
  else           { bop.x = -2.0f * sbz[scol]; bop.y = sbq[scol]; }

  // Fill d^2 slab: wave w handles n-tiles w, w+8, ... (scalar loop bounds ->
  // EXEC all 1s around every WMMA).
  for (int t = wid; t < N_ / 16; t += 8) {
    const int m = t * 16 + (lane & 15);    // A-fragment point row
    // A operand (16x4 f32): lanes 0-15 {K0=x, K1=y}; lanes 16-31 {K2=z, K3=1}
    v2f aop;
    if (lane < 16) { aop.x = p[3 * m + 0]; aop.y = p[3 * m + 1]; }
    else           { aop.x = p[3 * m + 2]; aop.y = 1.0f; }

    v8f acc = {};
    acc = __builtin_amdgcn_wmma_f32_16x16x4_f32(
        /*neg_a=*/false, aop, /*neg_b=*/false, bop,
        /*c_mod=*/(short)0, acc, /*reuse_a=*/false, /*reuse_b=*/false);

    // D layout: VGPR v, lanes 0-15 -> point M=v, lanes 16-31 -> M=v+8;
    // seed N = lane&15.  acc = -2 p.s + |s|^2 ; add |p|^2.
    const int srow = lane & 15;
    const int nbase = t * 16 + ((lane >> 4) << 3);
#pragma unroll
    for (int v = 0; v < 8; ++v) {
      const int n = nbase + v;
      dist[srow][n] = acc[v] + psq[n];
    }
  }
  __syncthreads();

  // Top-(K+1) smallest per seed via 11 argmin-extract passes (wave-owned rows,
  // scalar loop over seeds -> no exec masking in the scan kernel).
  float acc_loss = 0.0f;
  for (int srow = wid; srow < nvalid; srow += 8) {
    float* row = dist[srow];
    for (int k = 0; k <= KNN; ++k) {
      float mv = BIGF; int mi = 0;
      for (int n = lane; n < N_; n += 32) {   // conflict-free row scan
        const float v = row[n];
        if (v < mv || (v == mv && n < mi)) { mv = v; mi = n; }
      }
      for (int off = 16; off > 0; off >>= 1) {
        const float ov = __shfl_xor(mv, off, 32);
        const int   oi = __shfl_xor(mi, off, 32);
        if (ov < mv || (ov == mv && oi < mi)) { mv = ov; mi = oi; }
      }
      if (lane == 0) row[mi] = BIGF;          // extract
      if (k > 0 && lane == 0) {               // drop self (k==0)
        const float d2 = mv;
        const float d  = (d2 > 0.0f) ? sqrtf(d2) : 0.0f;
        acc_loss += -d * expf(-d2 * HINV2);
      }
      // Wave-local LDS RAW ordering for the extraction mark.
      asm volatile("s_wait_dscnt 0" ::: "memory");
      __builtin_amdgcn_wave_barrier();
    }
  }
  if (lane == 0) {
    atomicAdd(out, acc_loss * (1.0f / (float)B_));
  }
}

// ---------------------------------------------------------------------------
extern "C" void kernel_launch(void* const* d_in, const int* in_sizes, int n_in,
                              void* d_out, int out_size, void* d_ws, size_t ws_size,
                              hipStream_t stream) {
  (void)in_sizes; (void)n_in; (void)out_size; (void)ws_size;
  const float* pcs = (const float*)d_in[0];          // [512, 2048, 3] f32
  float* out   = (float*)d_out;                      // scalar f32
  float* seeds = (float*)d_ws;                       // [512, 20, 3] f32 scratch

  krl_zero_out<<<1, 1, 0, stream>>>(out);
  krl_fps_kernel<<<B_, 256, 0, stream>>>(pcs, seeds);
  krl_knn_repulsion_kernel<<<B_ * 2, 256, 0, stream>>>(pcs, seeds, out);
}